// ModelNew_33921651704476
// MI455X (gfx1250) — compile-verified
//
#include <hip/hip_runtime.h>
#include <hip/hip_bf16.h>

// ---------------------------------------------------------------------------
// Chunked SSM state recurrence, rewritten as a weighted GEMM-reduction:
//   final[b,h,p,n] = exp(sum A) * init
//                  + sum_t X[t,p] * B[t,n] * exp( sum_{t'>t} A[t'] )
// HBM-bound: ~340 MB mandatory reads @ 23.3 TB/s ~ 15 us floor. GEMM uses
// V_WMMA_F32_16X16X4_F32 (exact f32; matrix pipe is free at this intensity).
// This revision removes A-read amplification (transpose A to (b,h,s) once)
// and cuts per-chunk barriers 14 -> 2 via a single-wave shuffle scan.
// ---------------------------------------------------------------------------

typedef __attribute__((ext_vector_type(2))) float v2f;
typedef __attribute__((ext_vector_type(8))) float v8f;

#define BSZ 16
#define SEQ 8192
#define NH  8
#define PP  64
#define NN  16
#define LCH 64
#define NCH (SEQ / LCH)   // 128 chunks
#define NG  8             // chunk groups per (b,h)
#define CPG (NCH / NG)    // 16 chunks per group
#define BH  (BSZ * NH)    // 128

// workspace layout in floats (~8.5 MB total)
#define WS_AT   0                                   // At(b,h,s): BH*SEQ
#define WS_SUFF (BH * SEQ)                          // BH*NCH suffix sums
#define WS_ATOT (BH * SEQ + BH * NCH)               // BH totals
#define WS_PART (BH * SEQ + BH * NCH + BH)          // BH*NG*PP*NN partials

// ---------------------------------------------------------------------------
// Kernel 0: transpose A (b,s,h) -> At (b,h,s). Coalesced read and write via
// a padded LDS tile (stride 132 -> conflict-free both directions).
// ---------------------------------------------------------------------------
__global__ __launch_bounds__(128) void k_transposeA(const float* __restrict__ A,
                                                    float* __restrict__ ws) {
  __shared__ float t[NH * 132];
  const int tid = threadIdx.x;
  const int b  = blockIdx.x;            // 16
  const int st = blockIdx.y;            // 64 tiles of 128 s-values
  const size_t inbase = ((size_t)b * SEQ + (size_t)st * 128) * NH;
  float* At = ws + WS_AT;

#pragma unroll
  for (int i = 0; i < 8; ++i) {         // 1024 elements, coalesced read
    const int idx = tid + i * 128;      // flat (s,h)
    const int s = idx >> 3, h = idx & 7;
    t[h * 132 + s] = A[inbase + idx];
  }
  __syncthreads();
#pragma unroll
  for (int i = 0; i < 8; ++i) {         // coalesced write per h-row
    const int idx = tid + i * 128;
    const int h = idx >> 7, s = idx & 127;
    At[(size_t)(b * NH + h) * SEQ + (size_t)st * 128 + s] = t[h * 132 + s];
  }
}

// ---------------------------------------------------------------------------
// Kernel 1: per (b,h): chunk sums of At + suffix scan.
//   suffixAfter[bh][c] = sum of chunk sums for chunks > c ; Atot[bh] = total
// ---------------------------------------------------------------------------
__global__ __launch_bounds__(128) void k_chunksum(float* __restrict__ ws) {
  __shared__ float s[NCH];
  const int bh = blockIdx.x;
  const int c  = threadIdx.x;           // one chunk per thread
  const float* At = ws + WS_AT;

  // 256 B contiguous per thread, float4-vectorized
  const float4* ap = (const float4*)(At + (size_t)bh * SEQ + (size_t)c * LCH);
  float sum = 0.f;
#pragma unroll
  for (int i = 0; i < LCH / 4; ++i) {
    float4 v = ap[i];
    sum += (v.x + v.y) + (v.z + v.w);
  }
  s[c] = sum;
  __syncthreads();

  const float mine = sum;
  for (int off = 1; off < NCH; off <<= 1) {   // inclusive suffix scan
    float v = (c + off < NCH) ? s[c + off] : 0.f;
    __syncthreads();
    s[c] += v;
    __syncthreads();
  }
  ws[WS_SUFF + bh * NCH + c] = s[c] - mine;   // exclusive suffix
  if (c == 0) ws[WS_ATOT + bh] = s[0];        // total
}

// ---------------------------------------------------------------------------
// Kernel 2: each block = (b,h, group of 16 chunks), 128 threads = 4 waves.
// Per chunk: wave0 shuffle-scan -> weights in LDS (1 barrier) -> all waves
// stage bd = B*w (float4) in LDS (1 barrier) -> each wave computes one 16-row
// p-tile of partial[p,n] += X^T @ bd via 16 chained v_wmma_f32_16x16x4_f32.
// ---------------------------------------------------------------------------
__global__ __launch_bounds__(128) void k_chunk_gemm(const float* __restrict__ X,
                                                    const float* __restrict__ Bm,
                                                    float* __restrict__ ws) {
  __shared__ __align__(16) float lds_w[LCH];        // per-position weights
  __shared__ __align__(16) float lds_bd[LCH * NN];  // B * w, row stride 16

  const int bh   = blockIdx.x;
  const int g    = blockIdx.y;
  const int b    = bh / NH, h = bh % NH;
  const int tid  = threadIdx.x;
  const int lane = tid & 31;
  const int wave = tid >> 5;
  const int m    = lane & 15;           // row-in-tile (A) / col n (B,C)
  const int khi  = lane >> 4;           // lane-half selects K/M offset
  const int p0   = wave * 16;           // p-tile base for this wave

  const float* At   = ws + WS_AT + (size_t)bh * SEQ;
  const float* suff = ws + WS_SUFF + (size_t)bh * NCH;
  v8f acc = {};                          // 16x16 f32 accumulator tile

  for (int j = 0; j < CPG; ++j) {
    const int cc = g * CPG + j;
    const size_t s0     = (size_t)b * SEQ + (size_t)cc * LCH;
    const size_t xchunk = (s0 * NH + h) * (size_t)PP;   // + l*512 + p
    const size_t bbase  = (s0 * NH + h) * (size_t)NN;   // + l*128 + n

    // --- 1. wave0: coalesced float2 A-load, in-wave scan, weights ---------
    if (tid < 32) {
      const float2 a2 = ((const float2*)(At + (size_t)cc * LCH))[lane];
      const float ps = a2.x + a2.y;     // pair sum
      float sc = ps;                    // inclusive scan of pair sums
#pragma unroll
      for (int d = 1; d < 32; d <<= 1) {
        float tsh = __shfl_up(sc, d, 32);
        if (lane >= d) sc += tsh;
      }
      const float total = __shfl(sc, 31, 32);   // chunk sum S_c
      const float excl  = sc - ps;
      const float e     = suff[cc] + total;     // suffixAfter + S_c
      float2 wv;
      wv.x = __expf(e - (excl + a2.x));         // exp(sum_{t'>t} a)
      wv.y = __expf(e - (excl + a2.x + a2.y));
      ((float2*)lds_w)[lane] = wv;
    }
    __syncthreads();

    // --- 2. bd[l][n] = B[l][n] * w[l] into LDS (float4, coalesced) --------
#pragma unroll
    for (int i = 0; i < 2; ++i) {       // 256 float4s / 128 threads
      const int q  = tid + i * 128;
      const int l  = q >> 2, n4 = q & 3;
      float4 bv = ((const float4*)(Bm + bbase + (size_t)l * (NH * NN)))[n4];
      const float wl = lds_w[l];
      bv.x *= wl; bv.y *= wl; bv.z *= wl; bv.w *= wl;
      ((float4*)lds_bd)[q] = bv;
    }
    __syncthreads();

    // --- prefetch next chunk's X (128B line per thread) -------------------
    if (j + 1 < CPG) {
      const float* xn = X + xchunk + (size_t)LCH * (NH * PP)
                          + (size_t)(tid >> 1) * (NH * PP) + (tid & 1) * 32;
      __builtin_prefetch(xn, 0, 3);
    }

    // --- 3. GEMM: 16 k-steps of v_wmma_f32_16x16x4_f32 --------------------
    // f32 operand layout (ISA 7.12.2): K = vgpr + 2*(lane>=16)
    //   A lane (m,khi): a.x = X[l0+2khi][p0+m], a.y = X[l0+2khi+1][p0+m]
    //   B lane (n=m,khi): b.x = bd[l0+2khi][n], b.y = bd[l0+2khi+1][n]
    const size_t xlane = xchunk + (size_t)(2 * khi) * (NH * PP) + p0 + m;
    const int    bdl   = (2 * khi) * NN + m;
#pragma unroll
    for (int kk = 0; kk < 16; ++kk) {
      v2f a, bb;
      a.x  = X[xlane + (size_t)kk * 4 * (NH * PP)];
      a.y  = X[xlane + (size_t)kk * 4 * (NH * PP) + (NH * PP)];
      bb.x = lds_bd[bdl + kk * 4 * NN];
      bb.y = lds_bd[bdl + kk * 4 * NN + NN];
      acc = __builtin_amdgcn_wmma_f32_16x16x4_f32(
          /*neg_a=*/false, a, /*neg_b=*/false, bb,
          /*c_mod=*/(short)0, acc, /*reuse_a=*/false, /*reuse_b=*/false);
    }
    __syncthreads();   // next iteration overwrites lds_w/lds_bd
  }

  // --- write group partial: C/D layout M = r + 8*khi, N = m ---------------
  float* part = ws + WS_PART + ((size_t)bh * NG + g) * (PP * NN);
#pragma unroll
  for (int r = 0; r < 8; ++r) {
    const int p = p0 + r + 8 * khi;
    part[p * NN + m] = acc[r];
  }
}

// ---------------------------------------------------------------------------
// Kernel 3: out = exp(Atot) * init + sum_g partial_g   (fixed order -> det.)
// ---------------------------------------------------------------------------
__global__ __launch_bounds__(128) void k_finalize(const float* __restrict__ init,
                                                  const float* __restrict__ ws,
                                                  float* __restrict__ out) {
  const int bh  = blockIdx.x;
  const int tid = threadIdx.x;
  const float scale = __expf(ws[WS_ATOT + bh]);
  const float* part = ws + WS_PART + (size_t)bh * NG * (PP * NN);
  const size_t obase = (size_t)bh * (PP * NN);
#pragma unroll
  for (int i = 0; i < (PP * NN) / 128; ++i) {   // 8 elems/thread
    const int e = tid + i * 128;
    float a = scale * init[obase + e];
#pragma unroll
    for (int gg = 0; gg < NG; ++gg) a += part[(size_t)gg * (PP * NN) + e];
    out[obase + e] = a;
  }
}

// ---------------------------------------------------------------------------
extern "C" void kernel_launch(void* const* d_in, const int* in_sizes, int n_in,
                              void* d_out, int out_size, void* d_ws, size_t ws_size,
                              hipStream_t stream) {
  // setup_inputs order: X, initial_states, A, B   (all float32)
  const float* X    = (const float*)d_in[0];
  const float* init = (const float*)d_in[1];
  const float* A    = (const float*)d_in[2];
  const float* Bm   = (const float*)d_in[3];
  float* ws  = (float*)d_ws;   // needs ~8.5 MB
  float* out = (float*)d_out;

  k_transposeA<<<dim3(BSZ, SEQ / 128), 128, 0, stream>>>(A, ws);
  k_chunksum  <<<BH, 128, 0, stream>>>(ws);
  k_chunk_gemm<<<dim3(BH, NG), 128, 0, stream>>>(X, Bm, ws);
  k_finalize  <<<BH, 128, 0, stream>>>(init, ws, out);
}